// WMSA_51505247814074
// MI455X (gfx1250) — compile-verified
//
#include <hip/hip_runtime.h>

typedef float v2f  __attribute__((ext_vector_type(2)));
typedef float v8f  __attribute__((ext_vector_type(8)));
typedef int   i32x4 __attribute__((ext_vector_type(4)));

#define AS1 __attribute__((address_space(1)))
#define AS3 __attribute__((address_space(3)))

#define BATCH  8
#define NVALID 8100
#define NPAD   8128
#define WN     127          // windows per batch
#define PWIN   64           // window size
#define CDIM   512
#define HEADS  8
#define DHEAD  64
#define SVALID 36           // P - pad = 64 - 28

// ---------------------------------------------------------------------------
// CDNA5 async global->LDS copies (ASYNCcnt) with synchronous fallback
// ---------------------------------------------------------------------------
#if __has_builtin(__builtin_amdgcn_global_load_async_to_lds_b128) && \
    __has_builtin(__builtin_amdgcn_global_load_async_to_lds_b32)
#define HAVE_ASYNC 1
#else
#define HAVE_ASYNC 0
#endif

#if HAVE_ASYNC
#if __has_builtin(__builtin_amdgcn_s_wait_asynccnt)
#define WAIT_ASYNC(n) __builtin_amdgcn_s_wait_asynccnt(n)
#else
#define WAIT_ASYNC(n) asm volatile("s_wait_asynccnt %0" ::"i"(n))
#endif
#else
#define WAIT_ASYNC(n) ((void)0)
#endif

static __device__ __forceinline__ void async_copy_b128(const float* g, float* l) {
#if HAVE_ASYNC
    __builtin_amdgcn_global_load_async_to_lds_b128(
        (AS1 i32x4*)g, (AS3 i32x4*)l, 0, 0);
#else
    *(float4*)l = *(const float4*)g;
#endif
}
static __device__ __forceinline__ void async_copy_b32(const float* g, float* l) {
#if HAVE_ASYNC
    __builtin_amdgcn_global_load_async_to_lds_b32(
        (AS1 int*)g, (AS3 int*)l, 0, 0);
#else
    *l = *g;
#endif
}

static __device__ __forceinline__ v8f wmma_f32(v2f a, v2f b, v8f c) {
    // V_WMMA_F32_16X16X4_F32 : C(16x16,f32) += A(16x4,f32) x B(4x16,f32)
    return __builtin_amdgcn_wmma_f32_16x16x4_f32(
        false, a, false, b, (short)0, c, false, false);
}

// ---------------------------------------------------------------------------
// Tiled GEMM: C[M,N] = A[M,K] @ W[K,N] + bias[N]
//   PAD_IN : A rows indexed in padded space (b*NPAD+n); n >= NVALID reads 0.
//   TRIM_OUT: only rows with n < NVALID stored, at (b*NVALID+n)*N.
// Block tile 128x64, K-step 32, 256 threads (8 waves as 4x2 of 32x32).
// Double-buffered LDS fed by async global->LDS copies.
// ---------------------------------------------------------------------------
template <bool PAD_IN, bool TRIM_OUT>
__global__ void __launch_bounds__(256)
gemm_kernel(const float* __restrict__ A, const float* __restrict__ W,
            const float* __restrict__ bias, float* __restrict__ C,
            int N, int K)
{
    constexpr int BM = 128, BN = 64, BK = 32, LDT = 36; // 16B-aligned rows
    __shared__ __align__(16) float As[2][BM][LDT];
    __shared__ __align__(16) float Bs[2][BN][LDT];      // Bs[buf][n][k]

    const int tid  = threadIdx.x;
    const int lane = tid & 31;
    const int wave = tid >> 5;
    const int half = lane >> 4;
    const int l    = lane & 15;

    const int mblk = blockIdx.y * BM;
    const int nblk = blockIdx.x * BN;
    const int m0   = (wave & 3) * 32;
    const int n0   = (wave >> 2) * 32;

    v8f acc[2][2];
    acc[0][0] = {}; acc[0][1] = {}; acc[1][0] = {}; acc[1][1] = {};

    const int ar = tid >> 3, ac = (tid & 7) * 4;  // A map: 4 x b128 / thread
    const int wk = tid >> 3, wn = (tid & 7) * 8;  // W map: 8 x b32  / thread

    const int  bidx   = PAD_IN ? (mblk / NPAD) : 0;
    const int  nfirst = PAD_IN ? (mblk - bidx * NPAD) : 0;
    const bool dirty  = PAD_IN && (nfirst + BM > NVALID); // touches pad rows

    auto compute = [&](int buf) {
#pragma unroll
        for (int kk = 0; kk < BK; kk += 4) {
            const v2f a0 = *(const v2f*)&As[buf][m0 + l][kk + 2 * half];
            const v2f a1 = *(const v2f*)&As[buf][m0 + 16 + l][kk + 2 * half];
            const v2f b0 = *(const v2f*)&Bs[buf][n0 + l][kk + 2 * half];
            const v2f b1 = *(const v2f*)&Bs[buf][n0 + 16 + l][kk + 2 * half];
            acc[0][0] = wmma_f32(a0, b0, acc[0][0]);
            acc[0][1] = wmma_f32(a0, b1, acc[0][1]);
            acc[1][0] = wmma_f32(a1, b0, acc[1][0]);
            acc[1][1] = wmma_f32(a1, b1, acc[1][1]);
        }
    };

    if (!dirty) {
        // all 128 rows valid and contiguous in memory
        const float* Abase = PAD_IN ? (A + ((size_t)bidx * NVALID + nfirst) * K)
                                    : (A + (size_t)mblk * K);
        auto issue = [&](int buf, int k0) {
#pragma unroll
            for (int rr = 0; rr < 4; ++rr) {
                const int r = ar + rr * 32;
                async_copy_b128(Abase + (size_t)r * K + k0 + ac, &As[buf][r][ac]);
            }
#pragma unroll
            for (int j = 0; j < 8; ++j)  // transpose via per-element async b32
                async_copy_b32(&W[(size_t)(k0 + wk) * N + nblk + wn + j],
                               &Bs[buf][wn + j][wk]);
        };
        issue(0, 0);
        const int niter = K / BK;
        for (int it = 0; it < niter; ++it) {
            const int cur = it & 1;
            if (it + 1 < niter) {
                issue(cur ^ 1, (it + 1) * BK); // prefetch next tile
                WAIT_ASYNC(12);                // 12 ops of next batch may remain
            } else {
                WAIT_ASYNC(0);
            }
            __syncthreads();
            compute(cur);
            __syncthreads();
        }
    } else {
        // slow path: per-row guarded synchronous loads (zero-fill pad rows)
        for (int k0 = 0; k0 < K; k0 += BK) {
#pragma unroll
            for (int rr = 0; rr < 4; ++rr) {
                const int r = ar + rr * 32, R = mblk + r;
                const int bb = R / NPAD, n = R - bb * NPAD;
                float4 val = make_float4(0.f, 0.f, 0.f, 0.f);
                if (n < NVALID)
                    val = *(const float4*)&A[((size_t)bb * NVALID + n) * K + k0 + ac];
                As[0][r][ac + 0] = val.x; As[0][r][ac + 1] = val.y;
                As[0][r][ac + 2] = val.z; As[0][r][ac + 3] = val.w;
            }
#pragma unroll
            for (int j = 0; j < 8; ++j)
                Bs[0][wn + j][wk] = W[(size_t)(k0 + wk) * N + nblk + wn + j];
            __syncthreads();
            compute(0);
            __syncthreads();
        }
    }

#pragma unroll
    for (int i = 0; i < 2; ++i) {
#pragma unroll
        for (int j = 0; j < 2; ++j) {
            const int n  = nblk + n0 + 16 * j + l;
            const float bv = bias[n];
#pragma unroll
            for (int v = 0; v < 8; ++v) {
                const int   m   = mblk + m0 + 16 * i + v + 8 * half;
                const float val = acc[i][j][v] + bv;
                if (TRIM_OUT) {
                    const int bb = m / NPAD, nn = m - bb * NPAD;
                    if (nn < NVALID)
                        C[((size_t)bb * NVALID + nn) * N + n] = val;
                } else {
                    C[(size_t)m * N + n] = val;
                }
            }
        }
    }
}

// ---------------------------------------------------------------------------
// Attention per (b, h, w): S = q k^T * scale + bias (+mask), softmax, O = P v
// qkv layout from GEMM1: row (b*WN+w)*64+p, col s*512 + h*64 + d  (s=0/1/2)
// ---------------------------------------------------------------------------
__global__ void __launch_bounds__(256)
attn_kernel(const float* __restrict__ qkv, const float* __restrict__ rel_pos,
            float* __restrict__ out)
{
    constexpr int LD = 68; // 16B-aligned rows, even for v2f frags
    __shared__ __align__(16) float Qs[PWIN][LD];
    __shared__ __align__(16) float Ks[PWIN][LD];
    __shared__ __align__(16) float Vt[PWIN][LD];   // Vt[d][p] = v[p][d]
    __shared__ __align__(16) float Ss[PWIN][LD];

    const int w  = blockIdx.x;
    const int h  = blockIdx.y;
    const int bz = blockIdx.z;

    const int tid  = threadIdx.x;
    const int lane = tid & 31;
    const int wave = tid >> 5;
    const int half = lane >> 4;
    const int l    = lane & 15;

    const size_t rowbase = ((size_t)(bz * WN + w)) * PWIN;
    const int    colq    = h * DHEAD;

    // stage q,k via async global->LDS; v via VGPR transpose
    const int lp  = tid >> 2;          // row p : 0..63
    const int ld4 = (tid & 3) * 16;    // col base: 0,16,32,48
    const float* row = qkv + (rowbase + lp) * (3 * CDIM) + colq;
#pragma unroll
    for (int j = 0; j < 4; ++j) {
        const int c = ld4 + 4 * j;
        async_copy_b128(&row[c], &Qs[lp][c]);
        async_copy_b128(&row[CDIM + c], &Ks[lp][c]);
    }
#pragma unroll
    for (int j = 0; j < 4; ++j) {
        const int c = ld4 + 4 * j;
        const float4 vv = *(const float4*)&row[2 * CDIM + c];
        Vt[c + 0][lp] = vv.x; Vt[c + 1][lp] = vv.y;
        Vt[c + 2][lp] = vv.z; Vt[c + 3][lp] = vv.w;
    }
    WAIT_ASYNC(0);
    __syncthreads();

    const int m0 = (wave & 3) * 16;   // 16 rows per wave
    const int n0 = (wave >> 2) * 32;  // 32 cols per wave

    // S = q @ k^T
    v8f s0 = {}, s1 = {};
#pragma unroll
    for (int kk = 0; kk < 64; kk += 4) {
        const v2f a  = *(const v2f*)&Qs[m0 + l][kk + 2 * half];
        const v2f b0 = *(const v2f*)&Ks[n0 + l][kk + 2 * half];
        const v2f b1 = *(const v2f*)&Ks[n0 + 16 + l][kk + 2 * half];
        s0 = wmma_f32(a, b0, s0);
        s1 = wmma_f32(a, b1, s1);
    }

    const float scale = 0.125f;   // D^-0.5, D=64
    const bool  lastw = (w == WN - 1);
#pragma unroll
    for (int j = 0; j < 2; ++j) {
        const v8f sv = j ? s1 : s0;
        const int n  = n0 + 16 * j + l;
#pragma unroll
        for (int v = 0; v < 8; ++v) {
            const int m = m0 + v + 8 * half;
            float val = sv[v] * scale
                      + rel_pos[h * (2 * PWIN - 1) + (m - n + PWIN - 1)];
            if (lastw && ((m < SVALID) != (n < SVALID))) val = -1e30f;
            Ss[m][n] = val;
        }
    }
    __syncthreads();

    // row softmax (threads 0..63, one row each)
    if (tid < PWIN) {
        float mx = -1e30f;
        for (int c = 0; c < PWIN; ++c) mx = fmaxf(mx, Ss[tid][c]);
        float sum = 0.f;
        for (int c = 0; c < PWIN; ++c) {
            const float e = __expf(Ss[tid][c] - mx);
            Ss[tid][c] = e;
            sum += e;
        }
        const float inv = 1.f / sum;
        for (int c = 0; c < PWIN; ++c) Ss[tid][c] *= inv;
    }
    __syncthreads();

    // O = P @ v   (B fragment from Vt[N][K])
    v8f o0 = {}, o1 = {};
#pragma unroll
    for (int kk = 0; kk < 64; kk += 4) {
        const v2f a  = *(const v2f*)&Ss[m0 + l][kk + 2 * half];
        const v2f b0 = *(const v2f*)&Vt[n0 + l][kk + 2 * half];
        const v2f b1 = *(const v2f*)&Vt[n0 + 16 + l][kk + 2 * half];
        o0 = wmma_f32(a, b0, o0);
        o1 = wmma_f32(a, b1, o1);
    }
#pragma unroll
    for (int j = 0; j < 2; ++j) {
        const v8f ov = j ? o1 : o0;
        const int n  = n0 + 16 * j + l;            // d within head
#pragma unroll
        for (int v = 0; v < 8; ++v) {
            const int m = m0 + v + 8 * half;        // p within window
            out[(rowbase + m) * CDIM + colq + n] = ov[v];
        }
    }
}

// ---------------------------------------------------------------------------
extern "C" void kernel_launch(void* const* d_in, const int* in_sizes, int n_in,
                              void* d_out, int out_size, void* d_ws, size_t ws_size,
                              hipStream_t stream)
{
    const float* x       = (const float*)d_in[0];
    const float* w_qkv   = (const float*)d_in[1];
    const float* b_qkv   = (const float*)d_in[2];
    const float* rel_pos = (const float*)d_in[3];
    const float* w_out   = (const float*)d_in[4];
    const float* b_out   = (const float*)d_in[5];
    float* out = (float*)d_out;

    float* qkv  = (float*)d_ws;                               // 65024 x 1536
    float* attn = qkv + (size_t)BATCH * NPAD * 3 * CDIM;      // 65024 x 512

    const int M = BATCH * NPAD;   // 65024 = 508 * 128 exactly
    dim3 blk(256);

    // 1) QKV projection (zero-padded input rows)
    gemm_kernel<true, false><<<dim3((3 * CDIM) / 64, M / 128), blk, 0, stream>>>(
        x, w_qkv, b_qkv, qkv, 3 * CDIM, CDIM);

    // 2) windowed attention
    attn_kernel<<<dim3(WN, HEADS, BATCH), blk, 0, stream>>>(qkv, rel_pos, attn);

    // 3) output projection (store only valid rows)
    gemm_kernel<false, true><<<dim3(CDIM / 64, M / 128), blk, 0, stream>>>(
        attn, w_out, b_out, out, CDIM, CDIM);
}